// LRISpatialGNN_3298534883898
// MI455X (gfx1250) — compile-verified
//
#include <hip/hip_runtime.h>

typedef __attribute__((ext_vector_type(2))) float    v2f;
typedef __attribute__((ext_vector_type(8))) float    v8f;
typedef __attribute__((ext_vector_type(4))) unsigned v4u;
typedef __attribute__((ext_vector_type(8))) int      v8i;
typedef __attribute__((ext_vector_type(4))) int      v4i;

#define NEG_SLOPE 0.2f
#define NNODES 50000
#define NEDGES 800000
#define NGRAPH 64

__device__ __forceinline__ float lrelu(float x) { return x > 0.f ? x : NEG_SLOPE * x; }

// Monotone float <-> uint mapping so unsigned atomicMax == float max (handles negatives).
__device__ __forceinline__ unsigned fkey(float f) {
    unsigned u = __float_as_uint(f);
    return (u & 0x80000000u) ? ~u : (u | 0x80000000u);
}
__device__ __forceinline__ float funkey(unsigned k) {
    return __uint_as_float((k & 0x80000000u) ? (k & 0x7fffffffu) : ~k);
}

// ---------------------------------------------------------------------------
// C[M,N] = A[M,K] @ B[K,N], row-major fp32, V_WMMA_F32_16X16X4_F32.
// Grid: x = groups of 8 M-tiles, y = N-tile. Each block stages its K x 16
// weight panel into LDS once via the Tensor Data Mover (TENSORcnt), then all
// 8 waves stream A from global and B from LDS.
//   A 16x4 frag : lane&15 = M row, (lane>>4)*2 + elem = K
//   B 4x16 frag : lane&15 = N col, (lane>>4)*2 + elem = K
//   C/D         : VGPR g -> M = g + (lane>>4)*8, N = lane&15
// ---------------------------------------------------------------------------
__global__ void gemm_wmma_f32(const float* __restrict__ A, const float* __restrict__ B,
                              float* __restrict__ C, int M, int N, int K) {
    extern __shared__ float Bs[];                // K x 16 panel (K*64 bytes)
    const int lane   = threadIdx.x & 31;
    const int waveId = threadIdx.x >> 5;
    const int tn     = blockIdx.y;
    const int tm     = blockIdx.x * (blockDim.x >> 5) + waveId;

    // ---- Stage the K x 16 panel of B (columns tn*16 .. tn*16+15) into LDS ----
#if __has_builtin(__builtin_amdgcn_tensor_load_to_lds) && \
    __has_builtin(__builtin_amdgcn_s_wait_tensorcnt)
    if (threadIdx.x == 0) {                      // TDM ignores EXEC: issues once
        unsigned lds_off = (unsigned)(unsigned long long)(void*)&Bs[0];
        unsigned long long ga = (unsigned long long)(const void*)(B + tn * 16);
        v4u g0;
        g0[0] = 1u;                                           // count=1 (valid user D#)
        g0[1] = lds_off;                                      // lds_addr (bytes)
        g0[2] = (unsigned)(ga & 0xffffffffu);                 // global_addr[31:0]
        g0[3] = (unsigned)((ga >> 32) & 0x01ffffffu)          // global_addr[56:32]
              | 0x80000000u;                                  // type = 2 ("image")
        v8i g1;
        unsigned long long st0 = (unsigned long long)(unsigned)N;  // dim0 stride (elems)
        g1[0] = (int)(2u << 16);                              // data_size = 4 bytes
        g1[1] = (int)((unsigned)N << 16);                     // tensor_dim0[15:0]
        g1[2] = (int)((((unsigned)N >> 16) & 0xffffu) | ((unsigned)K << 16)); // d0 hi | d1 lo
        g1[3] = (int)((((unsigned)K >> 16) & 0xffffu) | (16u << 16));         // d1 hi | tile_dim0=16
        g1[4] = (int)((unsigned)K & 0xffffu);                 // tile_dim1 = K, tile_dim2 = 0
        g1[5] = (int)(unsigned)(st0 & 0xffffffffu);           // tensor_dim0_stride lo
        g1[6] = (int)(unsigned)((st0 >> 32) & 0xffffu);       // stride0 hi | stride1 lo = 0
        g1[7] = 0;                                            // stride1 hi = 0 (2-D tile)
        v4i gz4 = {};                                         // groups 2/3 unused (2-D)
        v8i gz8 = {};                                         // extra group (6-arg form)
        __builtin_amdgcn_tensor_load_to_lds(g0, g1, gz4, gz4, gz8, 0);
        __builtin_amdgcn_s_wait_tensorcnt(0);
    }
#else
    for (int i = threadIdx.x; i < K * 16; i += blockDim.x)
        Bs[i] = B[(long)(i >> 4) * N + tn * 16 + (i & 15)];
#endif
    __syncthreads();

    if (tm >= (M >> 4)) return;                  // wave-uniform: EXEC stays all-ones

    const int r  = lane & 15;
    const int kh = (lane >> 4) << 1;             // 0 or 2: K sub-offset for this half-wave
    const float* Arow = A + (long)(tm * 16 + r) * K;

    v8f acc = {};
    for (int k0 = 0; k0 < K; k0 += 4) {
        const int ka = k0 + kh;
        v2f a, b;
        a.x = Arow[ka];
        a.y = Arow[ka + 1];
        b.x = Bs[ka * 16 + r];
        b.y = Bs[(ka + 1) * 16 + r];
        acc = __builtin_amdgcn_wmma_f32_16x16x4_f32(false, a, false, b,
                                                    (short)0, acc, false, false);
    }
    float* Cp = C + (long)(tm * 16 + (kh << 2)) * N + tn * 16 + r;  // kh<<2 = 0 or 8
#pragma unroll
    for (int g = 0; g < 8; ++g) Cp[(long)g * N] = acc[g];
}

// asrc/adst per (node, head); also seed segment-max with the self-loop logit.
__global__ void attn_prep(const float* __restrict__ h, const float* __restrict__ a_src,
                          const float* __restrict__ a_dst, float* __restrict__ asrc,
                          float* __restrict__ adst, unsigned* __restrict__ mkey,
                          int N, int H) {
    int t = blockIdx.x * blockDim.x + threadIdx.x;
    if (t >= N * H) return;
    int n = t / H, hd = t - n * H;
    const float* hp = h + ((long)n * H + hd) * 128;
    const float* as = a_src + hd * 128;
    const float* ad = a_dst + hd * 128;
    float s1 = 0.f, s2 = 0.f;
#pragma unroll 8
    for (int k = 0; k < 128; ++k) { float v = hp[k]; s1 += v * as[k]; s2 += v * ad[k]; }
    asrc[t] = s1;
    adst[t] = s2;
    mkey[t] = fkey(lrelu(s1 + s2));              // self-loop seeds the max
}

__global__ void fzero(float* __restrict__ p, long n) {
    long i = (long)blockIdx.x * blockDim.x + threadIdx.x;
    if (i < n) p[i] = 0.f;
}

__global__ void edge_max(const int* __restrict__ src, const int* __restrict__ dst,
                         const float* __restrict__ asrc, const float* __restrict__ adst,
                         float* __restrict__ ebuf, unsigned* __restrict__ mkey,
                         int E, int H) {
    int t = blockIdx.x * blockDim.x + threadIdx.x;
    if (t >= E * H) return;
    int e = t / H, hd = t - e * H;
    int s = src[e], d = dst[e];
    float v = lrelu(asrc[s * H + hd] + adst[d * H + hd]);
    ebuf[t] = v;
    atomicMax(&mkey[d * H + hd], fkey(v));
}

// z starts with the self-loop's exp term (plain store, runs after max pass).
__global__ void node_selfz(const float* __restrict__ asrc, const float* __restrict__ adst,
                           const unsigned* __restrict__ mkey, float* __restrict__ z, int NH) {
    int t = blockIdx.x * blockDim.x + threadIdx.x;
    if (t >= NH) return;
    z[t] = expf(lrelu(asrc[t] + adst[t]) - funkey(mkey[t]));
}

__global__ void edge_expsum(const int* __restrict__ dst, float* __restrict__ ebuf,
                            const unsigned* __restrict__ mkey, float* __restrict__ z,
                            int E, int H) {
    int t = blockIdx.x * blockDim.x + threadIdx.x;
    if (t >= E * H) return;
    int e = t / H, hd = t - e * H;
    int d = dst[e];
    float p = expf(ebuf[t] - funkey(mkey[d * H + hd]));
    ebuf[t] = p;
    atomicAdd(&z[d * H + hd], p);
}

// One wave per (edge, head): out[dst] += alpha * h[src], 4 atomics per lane.
__global__ void edge_scatter(const int* __restrict__ src, const int* __restrict__ dst,
                             const float* __restrict__ ebuf, const float* __restrict__ z,
                             const float* __restrict__ h, float* __restrict__ out,
                             int E, int H) {
    int lane = threadIdx.x & 31;
    long item = (long)blockIdx.x * (blockDim.x >> 5) + (threadIdx.x >> 5);
    if (item >= (long)E * H) return;
    int e = (int)(item / H), hd = (int)(item - (long)e * H);
    int s = src[e], d = dst[e];
    float alpha = ebuf[item] / z[d * H + hd];
    const float* hs = h + ((long)s * H + hd) * 128;
    float* op = out + ((long)d * H + hd) * 128;
#pragma unroll
    for (int k = lane; k < 128; k += 32) atomicAdd(&op[k], hs[k] * alpha);
}

// Self-loop message + bias (only kernel touching `out` now -> plain RMW is safe).
__global__ void node_finish(const float* __restrict__ h, const float* __restrict__ asrc,
                            const float* __restrict__ adst, const unsigned* __restrict__ mkey,
                            const float* __restrict__ z, const float* __restrict__ bias,
                            float* __restrict__ out, int N, int H) {
    long t = (long)blockIdx.x * blockDim.x + threadIdx.x;
    long total = (long)N * H * 128;
    if (t >= total) return;
    int k  = (int)(t & 127);
    long nh = t >> 7;                  // n*H + hd
    int hd = (int)(nh % H);
    float alpha = expf(lrelu(asrc[nh] + adst[nh]) - funkey(mkey[nh])) / z[nh];
    out[t] += h[t] * alpha + bias[hd * 128 + k];
}

__global__ void pool_zero(float* __restrict__ gsum, float* __restrict__ gcnt) {
    int t = blockIdx.x * blockDim.x + threadIdx.x;
    if (t < NGRAPH * 128) gsum[t] = 0.f;
    if (t < NGRAPH) gcnt[t] = 0.f;
}

__global__ void pool_acc(const float* __restrict__ h2, const int* __restrict__ batch,
                         float* __restrict__ gsum, float* __restrict__ gcnt, int N) {
    int lane = threadIdx.x & 31;
    long n = (long)blockIdx.x * (blockDim.x >> 5) + (threadIdx.x >> 5);
    if (n >= N) return;
    int g = batch[n];
#pragma unroll
    for (int k = lane; k < 128; k += 32) atomicAdd(&gsum[g * 128 + k], h2[n * 128 + k]);
    if (lane == 0) atomicAdd(&gcnt[g], 1.f);
}

__global__ void classifier(const float* __restrict__ gsum, const float* __restrict__ gcnt,
                           const float* __restrict__ Wc1, const float* __restrict__ bc1,
                           const float* __restrict__ Wc2, const float* __restrict__ bc2,
                           float* __restrict__ outp) {
    __shared__ float pooled[128];
    __shared__ float zl[128];
    int g = blockIdx.x, j = threadIdx.x;            // block = 128 threads
    float cnt = fmaxf(gcnt[g], 1.f);
    pooled[j] = gsum[g * 128 + j] / cnt;
    __syncthreads();
    float acc = bc1[j];
#pragma unroll 8
    for (int k = 0; k < 128; ++k) acc += pooled[k] * Wc1[k * 128 + j];
    zl[j] = fmaxf(acc, 0.f);
    __syncthreads();
    if (j < 10) {
        float acc2 = bc2[j];
#pragma unroll 8
        for (int k = 0; k < 128; ++k) acc2 += zl[k] * Wc2[k * 10 + j];
        outp[g * 10 + j] = acc2;
    }
}

static inline int cdiv(long a, long b) { return (int)((a + b - 1) / b); }

extern "C" void kernel_launch(void* const* d_in, const int* in_sizes, int n_in,
                              void* d_out, int out_size, void* d_ws, size_t ws_size,
                              hipStream_t stream) {
    const float* x      = (const float*)d_in[0];
    const int*   eidx   = (const int*)  d_in[1];
    const int*   batch  = (const int*)  d_in[2];
    const float* W1     = (const float*)d_in[3];
    const float* a_src1 = (const float*)d_in[4];
    const float* a_dst1 = (const float*)d_in[5];
    const float* b1     = (const float*)d_in[6];
    const float* W2     = (const float*)d_in[7];
    const float* a_src2 = (const float*)d_in[8];
    const float* a_dst2 = (const float*)d_in[9];
    const float* b2     = (const float*)d_in[10];
    const float* Wc1    = (const float*)d_in[11];
    const float* bc1    = (const float*)d_in[12];
    const float* Wc2    = (const float*)d_in[13];
    const float* bc2    = (const float*)d_in[14];
    float* out = (float*)d_out;

    const int* src = eidx;
    const int* dst = eidx + NEDGES;

    // Workspace layout (fp32 units). h-buffer reused for h2; out-buffer reused for out2.
    float* h1    = (float*)d_ws;                 // 50000*256  (layer2: h2 = 50000*128)
    float* out1  = h1   + (long)NNODES * 256;    // 50000*256  (layer2: out2 = 50000*128)
    float* ebuf  = out1 + (long)NNODES * 256;    // 800000*2
    float* asrcb = ebuf + (long)NEDGES * 2;      // 50000*2
    float* adstb = asrcb + NNODES * 2;           // 50000*2
    unsigned* mkeyb = (unsigned*)(adstb + NNODES * 2); // 50000*2
    float* zb    = (float*)mkeyb + NNODES * 2;   // 50000*2
    float* gsum  = zb + NNODES * 2;              // 64*128
    float* gcnt  = gsum + NGRAPH * 128;          // 64

    const int B = 256;
    const int mtiles = NNODES / 16;              // 3125
    const int mblocks = cdiv(mtiles, 8);         // 8 waves (M-tiles) per block

    // ---------------- Layer 1 (H = 2, D = 128) ----------------
    {
        const int H = 2;
        dim3 g1(mblocks, 256 / 16);
        gemm_wmma_f32<<<g1, B, 128 * 16 * sizeof(float), stream>>>(x, W1, h1,
                                                                   NNODES, 256, 128);
        attn_prep<<<cdiv((long)NNODES * H, B), B, 0, stream>>>(h1, a_src1, a_dst1,
                                                               asrcb, adstb, mkeyb, NNODES, H);
        fzero<<<cdiv((long)NNODES * H * 128, B), B, 0, stream>>>(out1, (long)NNODES * H * 128);
        edge_max<<<cdiv((long)NEDGES * H, B), B, 0, stream>>>(src, dst, asrcb, adstb,
                                                              ebuf, mkeyb, NEDGES, H);
        node_selfz<<<cdiv((long)NNODES * H, B), B, 0, stream>>>(asrcb, adstb, mkeyb, zb,
                                                                NNODES * H);
        edge_expsum<<<cdiv((long)NEDGES * H, B), B, 0, stream>>>(dst, ebuf, mkeyb, zb, NEDGES, H);
        edge_scatter<<<cdiv((long)NEDGES * H * 32, B), B, 0, stream>>>(src, dst, ebuf, zb,
                                                                       h1, out1, NEDGES, H);
        node_finish<<<cdiv((long)NNODES * H * 128, B), B, 0, stream>>>(h1, asrcb, adstb, mkeyb,
                                                                       zb, b1, out1, NNODES, H);
    }

    // ---------------- Layer 2 (H = 1, D = 128) ----------------
    float* h2   = h1;    // reuse (GEMM reads out1, writes h2 -> disjoint)
    float* out2 = out1;  // reuse (out1 dead after GEMM2)
    {
        const int H = 1;
        dim3 g2(mblocks, 128 / 16);
        gemm_wmma_f32<<<g2, B, 256 * 16 * sizeof(float), stream>>>(out1, W2, h2,
                                                                   NNODES, 128, 256);
        attn_prep<<<cdiv((long)NNODES * H, B), B, 0, stream>>>(h2, a_src2, a_dst2,
                                                               asrcb, adstb, mkeyb, NNODES, H);
        fzero<<<cdiv((long)NNODES * H * 128, B), B, 0, stream>>>(out2, (long)NNODES * H * 128);
        edge_max<<<cdiv((long)NEDGES * H, B), B, 0, stream>>>(src, dst, asrcb, adstb,
                                                              ebuf, mkeyb, NEDGES, H);
        node_selfz<<<cdiv((long)NNODES * H, B), B, 0, stream>>>(asrcb, adstb, mkeyb, zb,
                                                                NNODES * H);
        edge_expsum<<<cdiv((long)NEDGES * H, B), B, 0, stream>>>(dst, ebuf, mkeyb, zb, NEDGES, H);
        edge_scatter<<<cdiv((long)NEDGES * H * 32, B), B, 0, stream>>>(src, dst, ebuf, zb,
                                                                       h2, out2, NEDGES, H);
        node_finish<<<cdiv((long)NNODES * H * 128, B), B, 0, stream>>>(h2, asrcb, adstb, mkeyb,
                                                                       zb, b2, out2, NNODES, H);
    }

    // ---------------- Pool + classifier ----------------
    pool_zero<<<cdiv(NGRAPH * 128, B), B, 0, stream>>>(gsum, gcnt);
    pool_acc<<<cdiv((long)NNODES * 32, B), B, 0, stream>>>(out2, batch, gsum, gcnt, NNODES);
    classifier<<<NGRAPH, 128, 0, stream>>>(gsum, gcnt, Wc1, bc1, Wc2, bc2, out);
}